// krignn_73572789780693
// MI455X (gfx1250) — compile-verified
//
#include <hip/hip_runtime.h>
#include <cstddef>

// ---------------------------------------------------------------------------
// Problem constants (match reference)
// ---------------------------------------------------------------------------
constexpr int Bc = 512;   // batches
constexpr int N  = 256;   // keys / queries per batch
constexpr int Cc = 3;     // coord dim
constexpr int Hh = 128;   // MLP hidden
constexpr int Vv = 16;    // value dim
constexpr int NP = 272;   // padded kriging system size (257 -> 17*16)
constexpr int NBLK = NP / 16; // 17

typedef __attribute__((ext_vector_type(2))) float v2f;
typedef __attribute__((ext_vector_type(8))) float v8f;

// ---------------------------------------------------------------------------
// WMMA f32 16x16x16 macro-op built from 4x V_WMMA_F32_16X16X4_F32.
// A: 16x16 row-major (stride lda), B: 16x16 row-major (stride ldb).
// A-fragment (16x4 f32): lanes 0-15 hold M=0..15 {K=2h, 2h+1}, lanes 16-31
// the other K half. B-fragment (4x16): lanes hold N=lane%16, K split by half.
// f32 WMMA has no A/B NEG modifier (ISA: NEG={CNeg,0,0}) so negation for
// the Schur update C -= A*B is folded into the A-fragment load.
// ---------------------------------------------------------------------------
__device__ __forceinline__ v8f wmma_k16(const float* __restrict__ A, int lda,
                                        bool negA,
                                        const float* __restrict__ Bm, int ldb,
                                        v8f c) {
  const int l    = threadIdx.x & 31;
  const int half = l >> 4;
  const int mn   = l & 15;
  #pragma unroll
  for (int q = 0; q < 4; ++q) {
    const int k = 4 * q + 2 * half;
    v2f a, b;
    a.x = A[mn * lda + k];
    a.y = A[mn * lda + k + 1];
    if (negA) { a.x = -a.x; a.y = -a.y; }
    b.x = Bm[k * ldb + mn];
    b.y = Bm[(k + 1) * ldb + mn];
    c = __builtin_amdgcn_wmma_f32_16x16x4_f32(false, a, false, b,
                                              (short)0, c, false, false);
  }
  return c;
}

// C/D 16x16 f32 fragment: VGPR r holds row (r + 8*half), col = lane%16.
__device__ __forceinline__ v8f load_c_tile(const float* __restrict__ C, int ldc) {
  const int l = threadIdx.x & 31;
  const int half = l >> 4, n = l & 15;
  v8f c;
  #pragma unroll
  for (int r = 0; r < 8; ++r) c[r] = C[(r + 8 * half) * ldc + n];
  return c;
}
__device__ __forceinline__ void store_c_tile(float* __restrict__ C, int ldc, v8f c) {
  const int l = threadIdx.x & 31;
  const int half = l >> 4, n = l & 15;
  #pragma unroll
  for (int r = 0; r < 8; ++r) C[(r + 8 * half) * ldc + n] = c[r];
}

// ---------------------------------------------------------------------------
// Kernel 1: Ks = KEY * mlp(KEY), Qs = QUERY * mlp(QUERY).
// 32 points per block, 128 threads (4 waves). Hidden GEMM (128x128) via WMMA.
// ---------------------------------------------------------------------------
__global__ __launch_bounds__(128)
void mlp_scale_kernel(const float* __restrict__ KEY,
                      const float* __restrict__ QUERY,
                      const float* __restrict__ W1, const float* __restrict__ b1,
                      const float* __restrict__ W2, const float* __restrict__ b2,
                      const float* __restrict__ W3, const float* __restrict__ b3,
                      float* __restrict__ Ks, float* __restrict__ Qs) {
  constexpr int PTS = 32;
  constexpr int LDH = 132;                 // padded stride (4*33) -> clean banks
  __shared__ float Xs[PTS][Cc];
  __shared__ float h1[PTS][LDH];
  __shared__ float h2[PTS][LDH];

  const int tid = threadIdx.x;
  const int wid = tid >> 5;
  int bb = blockIdx.x;
  const int half_blocks = (Bc * N) / PTS;  // 4096 per tensor
  const float* src = KEY;
  float* dst = Ks;
  if (bb >= half_blocks) { bb -= half_blocks; src = QUERY; dst = Qs; }
  const long p0 = (long)bb * PTS;

  // stage input points
  for (int o = tid; o < PTS * Cc; o += 128)
    Xs[o / Cc][o % Cc] = src[p0 * Cc + o];
  __syncthreads();

  // layer 1: h1 = relu(X @ W1 + b1)   (K=3, VALU)
  for (int o = tid; o < PTS * Hh; o += 128) {
    const int pt = o >> 7, n = o & 127;
    float s = b1[n];
    #pragma unroll
    for (int c = 0; c < Cc; ++c) s += Xs[pt][c] * W1[c * Hh + n];
    h1[pt][n] = fmaxf(s, 0.0f);
  }
  __syncthreads();

  // layer 2: h2 = relu(h1 @ W2 + b2)  (WMMA f32 16x16x4, K=128)
  // 2 row-tiles x 8 col-tiles = 16 tiles over 4 waves.
  for (int t = wid; t < 16; t += 4) {
    const int ti = t >> 3, tj = t & 7;
    v8f c = {};
    #pragma unroll
    for (int kq = 0; kq < Hh / 16; ++kq)
      c = wmma_k16(&h1[ti * 16][kq * 16], LDH, false,
                   &W2[(kq * 16) * Hh + tj * 16], Hh, c);
    const int l = threadIdx.x & 31;
    const int halfl = l >> 4, n = l & 15;
    const int col = tj * 16 + n;
    #pragma unroll
    for (int r = 0; r < 8; ++r) {
      const int row = ti * 16 + r + 8 * halfl;
      h2[row][col] = fmaxf(c[r] + b2[col], 0.0f);
    }
  }
  __syncthreads();

  // layer 3 + coordinate scale: out = X * (h2 @ W3 + b3)
  for (int o = tid; o < PTS * Cc; o += 128) {
    const int pt = o / Cc, cc = o % Cc;
    float s = b3[cc];
    #pragma unroll 8
    for (int k = 0; k < Hh; ++k) s += h2[pt][k] * W3[k * Cc + cc];
    dst[(p0 + pt) * Cc + cc] = Xs[pt][cc] * s;
  }
}

// ---------------------------------------------------------------------------
// Kernel 2: build padded kriging matrix G (272x272) and RHS Vpad (272x16).
// Layout:  G[0:257,0:257] = [[1-exp(-d(Ks,Ks)), 1],[1^T, 0]], pad = identity.
// grid = nchunk*17 blocks of 256 threads; each block fills 16 rows.
// ---------------------------------------------------------------------------
__global__ __launch_bounds__(256)
void build_g_kernel(const float* __restrict__ Ks,
                    const float* __restrict__ VALUE,
                    float* __restrict__ G, float* __restrict__ X,
                    int base) {
  __shared__ float Kl[N][Cc];
  const int tid  = threadIdx.x;
  const int bloc = blockIdx.x / NBLK;
  const int tile = blockIdx.x % NBLK;
  const int batch = base + bloc;
  float* Gc = G + (size_t)bloc * NP * NP;
  float* Xc = X + (size_t)bloc * NP * Vv;

  for (int o = tid; o < N * Cc; o += 256)
    Kl[o / Cc][o % Cc] = Ks[((size_t)batch * N) * Cc + o];
  __syncthreads();

  const int i0 = tile * 16;
  for (int idx = tid; idx < 16 * NP; idx += 256) {
    const int r = idx / NP, j = idx % NP;
    const int i = i0 + r;
    float val;
    if (i < N + 1 && j < N + 1) {
      if (i == N || j == N) {
        val = (i == N && j == N) ? 0.0f : 1.0f;
      } else {
        float d2 = 0.0f;
        #pragma unroll
        for (int c = 0; c < Cc; ++c) {
          const float d = Kl[i][c] - Kl[j][c];
          d2 += d * d;
        }
        val = (d2 > 0.0f) ? (1.0f - __expf(-sqrtf(d2))) : 0.0f;
      }
    } else {
      val = (i == j) ? 1.0f : 0.0f;       // identity padding
    }
    Gc[(size_t)i * NP + j] = val;
  }
  // RHS rows for this tile
  for (int idx = tid; idx < 16 * Vv; idx += 256) {
    const int r = idx / Vv, c = idx % Vv;
    const int i = i0 + r;
    Xc[(size_t)i * Vv + c] =
        (i < N) ? VALUE[((size_t)batch * N + i) * Vv + c] : 0.0f;
  }
}

// ---------------------------------------------------------------------------
// Kernel 3: per-batch blocked LU with partial pivoting (NB=16) + 16-RHS solve.
// One workgroup (512 threads, 16 waves) per batch; G in global/L2, panel and
// U12 row-block staged in LDS; trailing Schur updates via WMMA f32 16x16x4.
// ---------------------------------------------------------------------------
__global__ __launch_bounds__(512)
void lu_solve_kernel(float* __restrict__ G, float* __restrict__ X) {
  __shared__ float P[NP][17];        // panel (global-row indexed), stride 17
  __shared__ float U12[16][273];     // block row of U
  __shared__ float xs[NP][17];       // 16 RHS
  __shared__ int   pivs[NP];
  __shared__ float redv[16];
  __shared__ int   redi[16];

  const int tid  = threadIdx.x;
  const int lane = tid & 31;
  const int wid  = tid >> 5;
  float* Gc = G + (size_t)blockIdx.x * NP * NP;
  float* Xc = X + (size_t)blockIdx.x * NP * Vv;

  // ---------------- factorization ----------------
  for (int kb = 0; kb < NBLK; ++kb) {
    const int k0  = kb * 16;
    const int rem = NP - k0;

    // stage panel (rows k0..NP-1, cols k0..k0+15)
    for (int idx = tid; idx < rem * 16; idx += 512) {
      const int r = k0 + idx / 16, c = idx % 16;
      P[r][c] = Gc[(size_t)r * NP + k0 + c];
    }
    __syncthreads();

    // panel factorization with partial pivoting
    for (int c = 0; c < 16; ++c) {
      const int j = k0 + c;
      // pivot search over column c, rows j..NP-1
      float bm = -1.0f; int br = j;
      for (int r = j + tid; r < NP; r += 512) {
        const float a = fabsf(P[r][c]);
        if (a > bm) { bm = a; br = r; }
      }
      #pragma unroll
      for (int off = 16; off > 0; off >>= 1) {
        const float om = __shfl_down(bm, off);
        const int   orr = __shfl_down(br, off);
        if (om > bm) { bm = om; br = orr; }
      }
      if (lane == 0) { redv[wid] = bm; redi[wid] = br; }
      __syncthreads();
      if (tid == 0) {
        float m = redv[0]; int ri = redi[0];
        for (int w = 1; w < 16; ++w)
          if (redv[w] > m) { m = redv[w]; ri = redi[w]; }
        pivs[j] = ri;
      }
      __syncthreads();
      const int pr = pivs[j];
      if (pr != j) {
        if (tid < 16) {                      // swap LDS panel rows
          const float t = P[j][tid]; P[j][tid] = P[pr][tid]; P[pr][tid] = t;
        }
        for (int cc = tid; cc < NP; cc += 512) {   // swap full global rows
          const float t = Gc[(size_t)j * NP + cc];
          Gc[(size_t)j * NP + cc] = Gc[(size_t)pr * NP + cc];
          Gc[(size_t)pr * NP + cc] = t;
        }
      }
      __threadfence();
      __syncthreads();
      // scale column
      const float dv = P[j][c];
      const float inv = (dv != 0.0f) ? 1.0f / dv : 0.0f;
      for (int r = j + 1 + tid; r < NP; r += 512) P[r][c] *= inv;
      __syncthreads();
      // rank-1 update of remaining panel columns
      const int nr = NP - 1 - j, ncc = 15 - c;
      for (int idx = tid; idx < nr * ncc; idx += 512) {
        const int r  = j + 1 + idx / ncc;
        const int cc = c + 1 + idx % ncc;
        P[r][cc] -= P[r][c] * P[j][cc];
      }
      __syncthreads();
    }

    // write panel back
    for (int idx = tid; idx < rem * 16; idx += 512) {
      const int r = k0 + idx / 16, c = idx % 16;
      Gc[(size_t)r * NP + k0 + c] = P[r][c];
    }

    const int tcols = NP - k0 - 16;
    // U12 = L11^{-1} * A12 : one thread per trailing column
    for (int j = k0 + 16 + tid; j < NP; j += 512) {
      float u[16];
      #pragma unroll
      for (int r = 0; r < 16; ++r) {
        float s = Gc[(size_t)(k0 + r) * NP + j];
        for (int p = 0; p < r; ++p) s -= P[k0 + r][p] * u[p];
        u[r] = s;
        U12[r][j - (k0 + 16)] = s;
        Gc[(size_t)(k0 + r) * NP + j] = s;
      }
    }
    __threadfence();
    __syncthreads();

    // trailing update  A22 -= L21 * U12   (WMMA, one 16x16 tile per wave)
    const int t = tcols / 16;
    for (int tile = wid; tile < t * t; tile += 16) {
      const int ti = tile / t, tj = tile % t;
      const int i0 = k0 + 16 + ti * 16;
      float* Cptr = &Gc[(size_t)i0 * NP + (k0 + 16 + tj * 16)];
      v8f c = load_c_tile(Cptr, NP);
      c = wmma_k16(&P[i0][0], 17, /*negA=*/true, &U12[0][tj * 16], 273, c);
      store_c_tile(Cptr, NP, c);
    }
    __threadfence();
    __syncthreads();
  }

  // ---------------- 16-RHS solve: x = U^{-1} L^{-1} P * Vpad ----------------
  for (int idx = tid; idx < NP * Vv; idx += 512)
    xs[idx / Vv][idx % Vv] = Xc[idx];
  __syncthreads();

  if (wid == 0 && lane < Vv) {           // apply pivots, lane = RHS column
    for (int j = 0; j < NP; ++j) {
      const int pr = pivs[j];
      if (pr != j) { const float t = xs[j][lane]; xs[j][lane] = xs[pr][lane]; xs[pr][lane] = t; }
    }
  }
  __syncthreads();

  // forward (unit lower)
  for (int kb = 0; kb < NBLK; ++kb) {
    const int k0 = kb * 16;
    if (wid == 0 && lane < Vv) {
      for (int r = 1; r < 16; ++r) {
        float s = xs[k0 + r][lane];
        for (int p = 0; p < r; ++p)
          s -= Gc[(size_t)(k0 + r) * NP + k0 + p] * xs[k0 + p][lane];
        xs[k0 + r][lane] = s;
      }
    }
    __syncthreads();
    const int nrows = NP - k0 - 16;
    for (int idx = tid; idx < nrows * Vv; idx += 512) {
      const int i = k0 + 16 + idx / Vv, cc = idx % Vv;
      float s = xs[i][cc];
      #pragma unroll
      for (int p = 0; p < 16; ++p)
        s -= Gc[(size_t)i * NP + k0 + p] * xs[k0 + p][cc];
      xs[i][cc] = s;
    }
    __syncthreads();
  }

  // backward (upper)
  for (int kb = NBLK - 1; kb >= 0; --kb) {
    const int k0 = kb * 16;
    if (wid == 0 && lane < Vv) {
      for (int r = 15; r >= 0; --r) {
        float s = xs[k0 + r][lane];
        for (int p = r + 1; p < 16; ++p)
          s -= Gc[(size_t)(k0 + r) * NP + k0 + p] * xs[k0 + p][lane];
        const float d = Gc[(size_t)(k0 + r) * NP + k0 + r];
        xs[k0 + r][lane] = (d != 0.0f) ? s / d : 0.0f;
      }
    }
    __syncthreads();
    for (int idx = tid; idx < k0 * Vv; idx += 512) {
      const int i = idx / Vv, cc = idx % Vv;
      float s = xs[i][cc];
      #pragma unroll
      for (int p = 0; p < 16; ++p)
        s -= Gc[(size_t)i * NP + k0 + p] * xs[k0 + p][cc];
      xs[i][cc] = s;
    }
    __syncthreads();
  }

  // only rows 0..255 are needed downstream (Gs row 256 is zero)
  for (int idx = tid; idx < N * Vv; idx += 512)
    Xc[idx] = xs[idx / Vv][idx % Vv];
}

// ---------------------------------------------------------------------------
// Kernel 4: out[b,j,k] = sum_i (1-exp(-||Ks_i - Qs_j||)) * x[b,i,k]
// one block (256 threads) per batch; thread j owns one query row.
// ---------------------------------------------------------------------------
__global__ __launch_bounds__(256)
void out_kernel(const float* __restrict__ Ks, const float* __restrict__ Qs,
                const float* __restrict__ X, float* __restrict__ out, int base) {
  __shared__ float xl[N][17];
  __shared__ float Kl[N][Cc];
  const int tid = threadIdx.x;
  const int batch = base + blockIdx.x;
  const float* Xc = X + (size_t)blockIdx.x * NP * Vv;

  for (int o = tid; o < N * Vv; o += 256) xl[o / Vv][o % Vv] = Xc[o];
  for (int o = tid; o < N * Cc; o += 256)
    Kl[o / Cc][o % Cc] = Ks[((size_t)batch * N) * Cc + o];
  __syncthreads();

  const int j = tid;
  const float q0 = Qs[((size_t)batch * N + j) * Cc + 0];
  const float q1 = Qs[((size_t)batch * N + j) * Cc + 1];
  const float q2 = Qs[((size_t)batch * N + j) * Cc + 2];
  float acc[Vv];
  #pragma unroll
  for (int k = 0; k < Vv; ++k) acc[k] = 0.0f;

  for (int i = 0; i < N; ++i) {
    const float d0 = Kl[i][0] - q0;
    const float d1 = Kl[i][1] - q1;
    const float d2v = Kl[i][2] - q2;
    const float d2 = d0 * d0 + d1 * d1 + d2v * d2v;
    const float g = (d2 > 0.0f) ? (1.0f - __expf(-sqrtf(d2))) : 0.0f;
    #pragma unroll
    for (int k = 0; k < Vv; ++k) acc[k] += g * xl[i][k];
  }
  #pragma unroll
  for (int k = 0; k < Vv; ++k)
    out[((size_t)batch * N + j) * Vv + k] = acc[k];
}

// ---------------------------------------------------------------------------
// Host launcher
// ---------------------------------------------------------------------------
extern "C" void kernel_launch(void* const* d_in, const int* in_sizes, int n_in,
                              void* d_out, int out_size, void* d_ws, size_t ws_size,
                              hipStream_t stream) {
  const float* KEY   = (const float*)d_in[0];
  const float* VALUE = (const float*)d_in[1];
  const float* QUERY = (const float*)d_in[2];
  const float* W1 = (const float*)d_in[3];
  const float* b1 = (const float*)d_in[4];
  const float* W2 = (const float*)d_in[5];
  const float* b2 = (const float*)d_in[6];
  const float* W3 = (const float*)d_in[7];
  const float* b3 = (const float*)d_in[8];
  float* out = (float*)d_out;
  float* ws  = (float*)d_ws;

  const size_t ksElems = (size_t)Bc * N * Cc;
  float* Ks = ws;
  float* Qs = ws + ksElems;
  float* chunkBase = Qs + ksElems;

  // chunk the batched solve by available workspace (deterministic in ws_size)
  const size_t perBatch = (size_t)NP * NP + (size_t)NP * Vv;   // floats
  const size_t totalFloats = ws_size / sizeof(float);
  size_t avail = (totalFloats > 2 * ksElems) ? (totalFloats - 2 * ksElems) : 0;
  int chunk = (int)(avail / perBatch);
  if (chunk < 1) chunk = 1;
  if (chunk > Bc) chunk = Bc;

  float* G = chunkBase;
  float* X = G + (size_t)chunk * NP * NP;

  // 1) coordinate-rescaling MLP for all keys and queries
  {
    const int blocks = 2 * (Bc * N) / 32;   // 8192
    mlp_scale_kernel<<<blocks, 128, 0, stream>>>(KEY, QUERY, W1, b1, W2, b2,
                                                 W3, b3, Ks, Qs);
  }

  // 2..4) per-chunk: build system, LU+solve, output GEMV
  for (int base = 0; base < Bc; base += chunk) {
    const int nb = (base + chunk <= Bc) ? chunk : (Bc - base);
    build_g_kernel<<<nb * NBLK, 256, 0, stream>>>(Ks, VALUE, G, X, base);
    lu_solve_kernel<<<nb, 512, 0, stream>>>(G, X);
    out_kernel<<<nb, 256, 0, stream>>>(Ks, Qs, X, out, base);
  }
}